// PeptideGCN_7095285973785
// MI455X (gfx1250) — compile-verified
//
#include <hip/hip_runtime.h>

typedef __attribute__((ext_vector_type(2))) float v2f;
typedef __attribute__((ext_vector_type(8))) float v8f;

static constexpr int HDIM = 128;          // hidden width (fixed by reference)
#define BN_EPS 1e-5f

static inline int cdiv_i(long long a, long long b) { return (int)((a + b - 1) / b); }

// ---------------------------------------------------------------------------
// elementwise fill
__global__ void k_fill_f32(float* __restrict__ p, float v, long long n) {
  long long i = (long long)blockIdx.x * blockDim.x + threadIdx.x;
  if (i < n) p[i] = v;
}

// deg[dst] += w  (deg pre-filled with 1.0 for the self-loop)
__global__ void k_deg(const int* __restrict__ dst, const float* __restrict__ w,
                      float* __restrict__ deg, int E) {
  int e = blockIdx.x * blockDim.x + threadIdx.x;
  if (e < E) atomicAdd(&deg[dst[e]], w[e]);
}

__global__ void k_node_norm(const float* __restrict__ deg, float* __restrict__ dis,
                            float* __restrict__ self_norm, int n) {
  int i = blockIdx.x * blockDim.x + threadIdx.x;
  if (i < n) { float d = deg[i]; dis[i] = rsqrtf(d); self_norm[i] = 1.0f / d; }
}

__global__ void k_edge_norm(const int* __restrict__ src, const int* __restrict__ dst,
                            const float* __restrict__ w, const float* __restrict__ dis,
                            float* __restrict__ norm, int E) {
  int e = blockIdx.x * blockDim.x + threadIdx.x;
  if (e < E) norm[e] = dis[src[e]] * w[e] * dis[dst[e]];
}

// ---------------------------------------------------------------------------
// C[M, NT*16] = A[M,K] @ B[K, NT*16] (+bias) (optional ReLU).
// V_WMMA_F32_16X16X4_F32, wave32 per 16-row strip, NT accumulator tiles per wave.
// B staged through LDS in 64-row chunks (<=32KB): cooperative b128 loads,
// all waves reach both barriers (out-of-range waves clamp, never early-return,
// keeping EXEC all-1s for WMMA).
template <int NT>
__global__ void __launch_bounds__(256)
k_gemm_wmma_f32(const float* __restrict__ A, const float* __restrict__ B,
                const float* __restrict__ bias, float* __restrict__ C,
                int M, int K, int relu) {
  constexpr int Ncol = NT * 16;
  constexpr int KC   = 64;                      // K-chunk staged in LDS
  __shared__ __align__(16) float ldsB[KC * Ncol];

  const int tid  = threadIdx.x;
  const int lane = tid & 31;
  const int wave = tid >> 5;                    // 8 waves -> 128 rows per block
  const int row0 = blockIdx.x * 128 + wave * 16;
  const int hi   = lane >> 4;                   // 0: lanes 0-15, 1: lanes 16-31

  int row = row0 + (lane & 15);
  if (row >= M) row = M - 1;                    // clamp: EXEC stays uniform
  const float* __restrict__ arow = A + (long long)row * K;

  v8f zero = {};
  v8f acc[NT];
#pragma unroll
  for (int t = 0; t < NT; ++t) acc[t] = zero;

  for (int kc = 0; kc < K; kc += KC) {
    const int klen = (K - kc) < KC ? (K - kc) : KC;   // multiple of 4 here
    __syncthreads();
    // cooperative stage of B[kc:kc+klen, :] into LDS (16B vectors)
    for (int i = tid * 4; i < klen * Ncol; i += 256 * 4)
      *(float4*)(ldsB + i) = *(const float4*)(B + (long long)kc * Ncol + i);
    __syncthreads();

    if (kc + KC < K) __builtin_prefetch(arow + kc + KC, 0, 0);  // global_prefetch_b8

    for (int k0 = 0; k0 < klen; k0 += 4) {
      const int k = k0 + 2 * hi;                // ISA 16x16x4 fragment layout
      v2f a;
      a.x = arow[kc + k];
      a.y = arow[kc + k + 1];
      const float* bp = ldsB + k * Ncol + (lane & 15);
#pragma unroll
      for (int t = 0; t < NT; ++t) {            // reuse A fragment across tiles
        v2f b;
        b.x = bp[t * 16];
        b.y = bp[t * 16 + Ncol];
        acc[t] = __builtin_amdgcn_wmma_f32_16x16x4_f32(
            /*neg_a=*/false, a, /*neg_b=*/false, b,
            /*c_mod=*/(short)0, acc[t], /*reuse_a=*/false, /*reuse_b=*/false);
      }
    }
  }

  const int m0 = row0 + 8 * hi;                 // C VGPR j -> row m0+j
#pragma unroll
  for (int t = 0; t < NT; ++t) {
    const int col = t * 16 + (lane & 15);
    const float bv = bias ? bias[col] : 0.0f;
#pragma unroll
    for (int j = 0; j < 8; ++j) {
      int m = m0 + j;
      if (m < M) {
        float v = acc[t][j] + bv;
        if (relu) v = fmaxf(v, 0.0f);
        C[(long long)m * Ncol + col] = v;
      }
    }
  }
}

// ---------------------------------------------------------------------------
// agg[dst[e]][f] += norm[e] * xw[src[e]][f]   (lane f-major => coalesced)
__global__ void k_scatter(const int* __restrict__ src, const int* __restrict__ dst,
                          const float* __restrict__ norm, const float* __restrict__ xw,
                          float* __restrict__ agg, long long total) {
  long long i = (long long)blockIdx.x * blockDim.x + threadIdx.x;
  if (i >= total) return;
  int e = (int)(i >> 7);                        // HDIM == 128
  int f = (int)(i & (HDIM - 1));
  float v = norm[e] * xw[(long long)src[e] * HDIM + f];
  atomicAdd(&agg[(long long)dst[e] * HDIM + f], v);
}

// out = relu(BN(agg + self_norm*xw + bias))
__global__ void k_gcn_epilogue(const float* __restrict__ agg, const float* __restrict__ xw,
                               const float* __restrict__ self_norm, const float* __restrict__ bias,
                               const float* __restrict__ bg, const float* __restrict__ bb,
                               const float* __restrict__ bm, const float* __restrict__ bv,
                               float* __restrict__ out, long long total) {
  long long i = (long long)blockIdx.x * blockDim.x + threadIdx.x;
  if (i >= total) return;
  int node = (int)(i >> 7);
  int f    = (int)(i & (HDIM - 1));
  float val = agg[i] + self_norm[node] * xw[i] + bias[f];
  val = (val - bm[f]) * rsqrtf(bv[f] + BN_EPS) * bg[f] + bb[f];
  out[i] = fmaxf(val, 0.0f);
}

// ---------------------------------------------------------------------------
__global__ void k_count(const int* __restrict__ batch, float* __restrict__ cnt, int n) {
  int i = blockIdx.x * blockDim.x + threadIdx.x;
  if (i < n) atomicAdd(&cnt[batch[i]], 1.0f);
}

// gcat[g][0:128]   = segment-max(h)  (h >= 0 so uint-bit atomicMax is exact; 0-init
//                    also matches where(counts>0, max, 0))
// gcat[g][128:256] = segment-sum(h)  (divided by count afterwards)
__global__ void k_pool(const float* __restrict__ h, const int* __restrict__ batch,
                       float* __restrict__ gcat, long long total) {
  long long i = (long long)blockIdx.x * blockDim.x + threadIdx.x;
  if (i >= total) return;
  int node = (int)(i >> 7);
  int f    = (int)(i & (HDIM - 1));
  float v  = h[i];                              // >= 0 after ReLU
  long long base = (long long)batch[node] * (2 * HDIM);
  atomicMax((unsigned int*)&gcat[base + f], __float_as_uint(v));
  atomicAdd(&gcat[base + HDIM + f], v);
}

__global__ void k_pool_mean(float* __restrict__ gcat, const float* __restrict__ cnt,
                            long long totalGH) {
  long long i = (long long)blockIdx.x * blockDim.x + threadIdx.x;
  if (i >= totalGH) return;
  int g = (int)(i >> 7);
  int f = (int)(i & (HDIM - 1));
  gcat[(long long)g * (2 * HDIM) + HDIM + f] /= fmaxf(cnt[g], 1.0f);
}

// out[G,2] = A[G,64] @ W[64,2] + b   (tiny head; scalar kernel)
__global__ void k_final(const float* __restrict__ A, const float* __restrict__ W,
                        const float* __restrict__ b, float* __restrict__ out, int G) {
  int i = blockIdx.x * blockDim.x + threadIdx.x;
  if (i >= G * 2) return;
  int g = i >> 1, o = i & 1;
  const float* a = A + (long long)g * 64;
  float s = b[o];
#pragma unroll
  for (int k = 0; k < 64; ++k) s += a[k] * W[k * 2 + o];
  out[i] = s;
}

// ---------------------------------------------------------------------------
extern "C" void kernel_launch(void* const* d_in, const int* in_sizes, int n_in,
                              void* d_out, int out_size, void* d_ws, size_t ws_size,
                              hipStream_t stream) {
  const float* x     = (const float*)d_in[0];
  const int*   eidx  = (const int*)d_in[1];      // [2,E] row-major: src=eidx, dst=eidx+E
  const float* ew    = (const float*)d_in[2];
  const int*   batch = (const int*)d_in[3];
  const float* W1 = (const float*)d_in[5];  const float* b1 = (const float*)d_in[6];
  const float* W2 = (const float*)d_in[7];  const float* b2 = (const float*)d_in[8];
  const float* W3 = (const float*)d_in[9];  const float* b3 = (const float*)d_in[10];
  const float* bng[3] = {(const float*)d_in[11], (const float*)d_in[15], (const float*)d_in[19]};
  const float* bnb[3] = {(const float*)d_in[12], (const float*)d_in[16], (const float*)d_in[20]};
  const float* bnm[3] = {(const float*)d_in[13], (const float*)d_in[17], (const float*)d_in[21]};
  const float* bnv[3] = {(const float*)d_in[14], (const float*)d_in[18], (const float*)d_in[22]};
  const float* Wf1 = (const float*)d_in[23]; const float* bf1 = (const float*)d_in[24];
  const float* Wf2 = (const float*)d_in[25]; const float* bf2 = (const float*)d_in[26];
  const float* Wf3 = (const float*)d_in[27]; const float* bf3 = (const float*)d_in[28];

  const int N = in_sizes[3];
  const int E = in_sizes[2];
  const int D = in_sizes[0] / N;
  const int G = out_size / 2;

  // workspace carve (256B aligned)
  char* p = (char*)d_ws;
  auto carve = [&](size_t bytes) -> float* {
    float* r = (float*)p;
    p += (bytes + 255) & ~(size_t)255;
    return r;
  };
  float* deg  = carve((size_t)N * 4);
  float* dis  = carve((size_t)N * 4);
  float* sn   = carve((size_t)N * 4);
  float* nrm  = carve((size_t)E * 4);
  float* xw   = carve((size_t)N * HDIM * 4);
  float* agg  = carve((size_t)N * HDIM * 4);
  float* h    = carve((size_t)N * HDIM * 4);
  float* cnt  = carve((size_t)G * 4);
  float* gcat = carve((size_t)G * 2 * HDIM * 4);
  float* m1   = carve((size_t)G * HDIM * 4);
  float* m2   = carve((size_t)G * 64 * 4);

  const int T = 256;
  const long long NH = (long long)N * HDIM;
  const long long EH = (long long)E * HDIM;

  // --- normalization terms (self-loop weight 1) ---
  k_fill_f32<<<cdiv_i(N, T), T, 0, stream>>>(deg, 1.0f, N);
  k_deg<<<cdiv_i(E, T), T, 0, stream>>>(eidx + E, ew, deg, E);
  k_node_norm<<<cdiv_i(N, T), T, 0, stream>>>(deg, dis, sn, N);
  k_edge_norm<<<cdiv_i(E, T), T, 0, stream>>>(eidx, eidx + E, ew, dis, nrm, E);

  // --- three GCN layers ---
  const float* hin = x;
  int K = D;
  const float* Ws[3] = {W1, W2, W3};
  const float* bs[3] = {b1, b2, b3};
  for (int L = 0; L < 3; ++L) {
    k_gemm_wmma_f32<8><<<cdiv_i(N, 128), 256, 0, stream>>>(hin, Ws[L], nullptr, xw,
                                                           N, K, /*relu=*/0);
    k_fill_f32<<<cdiv_i(NH, T), T, 0, stream>>>(agg, 0.0f, NH);
    k_scatter<<<cdiv_i(EH, T), T, 0, stream>>>(eidx, eidx + E, nrm, xw, agg, EH);
    k_gcn_epilogue<<<cdiv_i(NH, T), T, 0, stream>>>(agg, xw, sn, bs[L],
                                                    bng[L], bnb[L], bnm[L], bnv[L], h, NH);
    hin = h;
    K = HDIM;
  }

  // --- graph pooling: [max | mean] -> gcat[G, 256] ---
  k_fill_f32<<<cdiv_i(G, T), T, 0, stream>>>(cnt, 0.0f, G);
  k_fill_f32<<<cdiv_i((long long)G * 2 * HDIM, T), T, 0, stream>>>(gcat, 0.0f,
                                                                   (long long)G * 2 * HDIM);
  k_count<<<cdiv_i(N, T), T, 0, stream>>>(batch, cnt, N);
  k_pool<<<cdiv_i(NH, T), T, 0, stream>>>(h, batch, gcat, NH);
  k_pool_mean<<<cdiv_i((long long)G * HDIM, T), T, 0, stream>>>(gcat, cnt,
                                                                (long long)G * HDIM);

  // --- MLP head ---
  k_gemm_wmma_f32<8><<<cdiv_i(G, 128), 256, 0, stream>>>(gcat, Wf1, bf1, m1,
                                                         G, 2 * HDIM, /*relu=*/1);
  k_gemm_wmma_f32<4><<<cdiv_i(G, 128), 256, 0, stream>>>(m1, Wf2, bf2, m2,
                                                         G, HDIM, /*relu=*/1);
  k_final<<<cdiv_i(2 * G, T), T, 0, stream>>>(m2, Wf3, bf3, (float*)d_out, G);
}